// SNN_Autoencoder_54082228191383
// MI455X (gfx1250) — compile-verified
//
#include <hip/hip_runtime.h>

typedef __attribute__((ext_vector_type(2))) float v2f;
typedef __attribute__((ext_vector_type(8))) float v8f;

// C[M,N] = A[M,K] * B[K,N], all row-major fp32, via V_WMMA_F32_16X16X4_F32.
// Each wave computes a 16 x (16*WT) tile; one A fragment reused across the WT
// accumulators per k-step (K stepped by 4). Software-pipelined: next k-step's
// fragments are loaded before the current step's WMMAs issue, so loads overlap
// the matrix pipe instead of stalling it.
//
// WMMA f32 operand layouts (ISA 7.12.2, wave32):
//   A 16x4:  lane l (l<16): row=l, K=k0+{0,1} ; lane l+16: row=l, K=k0+{2,3}
//   B 4x16:  lane l (l<16): col=l, K=k0+{0,1} ; lane l+16: col=l, K=k0+{2,3}
//   C/D 16x16: VGPR r: lanes 0-15 -> (M=r, N=lane), lanes 16-31 -> (M=r+8, N=lane-16)
template <int WT>
__global__ __launch_bounds__(256) void gemm_f32_wmma(const float* __restrict__ A,
                                                     const float* __restrict__ B,
                                                     float* __restrict__ C,
                                                     int M, int N, int K) {
  const int lane = threadIdx.x & 31;
  const int waveInBlock = threadIdx.x >> 5;
  const int wave = blockIdx.x * (blockDim.x >> 5) + waveInBlock;

  const int groupsN = N / (16 * WT);
  const int tm = wave / groupsN;  // 16-row tile index (wave-uniform)
  const int gn = wave % groupsN;  // (16*WT)-col group index (wave-uniform)
  if (tm * 16 >= M) return;       // uniform per wave -> EXEC stays all-1s

  const int half = lane >> 4;  // 0: K pair {0,1}, 1: K pair {2,3}
  const int l15 = lane & 15;

  v8f acc[WT];
#pragma unroll
  for (int t = 0; t < WT; ++t) acc[t] = (v8f){0.f, 0.f, 0.f, 0.f, 0.f, 0.f, 0.f, 0.f};

  // Incremental pointers: Ap advances 4 floats/iter, Bp advances 4 rows/iter.
  const float* __restrict__ Ap = A + (size_t)(tm * 16 + l15) * K + 2 * half;
  const float* __restrict__ Bp = B + (size_t)(2 * half) * N + gn * 16 * WT + l15;

  // Prologue: fragments for k0 = 0.
  v2f a = *(const v2f*)Ap;
  v2f b[WT];
#pragma unroll
  for (int t = 0; t < WT; ++t) {
    b[t].x = Bp[t * 16];
    b[t].y = Bp[t * 16 + N];
  }

#pragma unroll 2
  for (int k0 = 0; k0 < K - 4; k0 += 4) {
    Ap += 4;
    Bp += (size_t)4 * N;
    // Issue next-step loads first (fill the memory pipe)...
    v2f a_nxt = *(const v2f*)Ap;
    v2f b_nxt[WT];
#pragma unroll
    for (int t = 0; t < WT; ++t) {
      b_nxt[t].x = Bp[t * 16];
      b_nxt[t].y = Bp[t * 16 + N];
    }
    // ...then compute on the already-resident fragments.
#pragma unroll
    for (int t = 0; t < WT; ++t)
      acc[t] = __builtin_amdgcn_wmma_f32_16x16x4_f32(false, a, false, b[t], (short)0,
                                                     acc[t], false, false);
    a = a_nxt;
#pragma unroll
    for (int t = 0; t < WT; ++t) b[t] = b_nxt[t];
  }
  // Epilogue: last k-step.
#pragma unroll
  for (int t = 0; t < WT; ++t)
    acc[t] = __builtin_amdgcn_wmma_f32_16x16x4_f32(false, a, false, b[t], (short)0,
                                                   acc[t], false, false);

  // Store D: lanes 0-15 cover consecutive columns -> coalesced 64B per row-half.
  const int crow0 = tm * 16 + half * 8;
#pragma unroll
  for (int t = 0; t < WT; ++t) {
    float* __restrict__ Cp = C + (size_t)crow0 * N + gn * 16 * WT + t * 16 + l15;
#pragma unroll
    for (int r = 0; r < 8; ++r) Cp[(size_t)r * N] = acc[t][r];
  }
}

extern "C" void kernel_launch(void* const* d_in, const int* in_sizes, int n_in,
                              void* d_out, int out_size, void* d_ws, size_t ws_size,
                              hipStream_t stream) {
  (void)in_sizes; (void)n_in; (void)d_ws; (void)ws_size; (void)out_size;

  const float* x     = (const float*)d_in[0];  // [2048, 1024]
  const float* enc_w = (const float*)d_in[1];  // [1024, 256]
  const float* dec_w = (const float*)d_in[2];  // [256, 1024]

  const int Bn = 2048, Dn = 1024, Hn = 256;
  float* out = (float*)d_out;
  float* m_d = out;                    // [2048, 1024] (first return value)
  float* m_e = out + (size_t)Bn * Dn;  // [2048, 256]  (second return value)

  constexpr int WT = 4;  // 4 x 16 = 64 output columns per wave

  // GEMM1: m_e = x @ enc_w   (M=2048, K=1024, N=256)
  {
    int waves = (Bn / 16) * (Hn / (16 * WT));  // 128 * 4 = 512
    int blocks = (waves * 32 + 255) / 256;     // 64
    gemm_f32_wmma<WT><<<blocks, 256, 0, stream>>>(x, enc_w, m_e, Bn, Hn, Dn);
  }
  // GEMM2: m_d = m_e @ dec_w (M=2048, K=256, N=1024); reads m_e written above
  {
    int waves = (Bn / 16) * (Dn / (16 * WT));  // 128 * 16 = 2048
    int blocks = (waves * 32 + 255) / 256;     // 256
    gemm_f32_wmma<WT><<<blocks, 256, 0, stream>>>(m_e, dec_w, m_d, Bn, Dn, Hn);
  }
}